// PenetLoss_60155311947791
// MI455X (gfx1250) — compile-verified
//
#include <hip/hip_runtime.h>
#include <hip/hip_bf16.h>
#include <math.h>

typedef __attribute__((ext_vector_type(2))) float v2f;
typedef __attribute__((ext_vector_type(8))) float v8f;

#define B_  8
#define V_  32768
#define J_  21
#define P_  205
#define NC_ 165

// ---- workspace layout (32-bit words). d_ws 256B aligned; float4 regions 16B aligned.
#define WS_POS4   0                       // 205*4  : template bone points (x,y,z,|p|^2)
#define WS_BONE4  820                     // 8*205*4: posed bone points (x,y,z,|b|^2)
#define WS_MINS   7380                    // 8*208 u: flipped min of (|v|^2 - 2 v.p)
#define WS_NRS    9044                    // 8*48  u: flipped masked min (non-rigid)
#define WS_RACC   9428                    // 8      : rigid accumulator
#define WS_MESH4  9440                    // 8*32768*4 : packed A operand (x,y,z,|v|^2)
#define WS_GEO4   (WS_MESH4 + B_*V_*4)    // 8*32768*4 : packed A operand (x,y,z,mask|g|^2)
#define WS_WORDS_PACKED (WS_GEO4 + B_*V_*4)

// ---------- compile-time combo table (segment base bone-indices) ----------
struct ComboTab { short s1[NC_]; short s2[NC_]; };
constexpr ComboTab makeCombos() {
  ComboTab t{};
  int n = 0;
  for (int i = 0; i < 20; ++i) {
    int f1 = i / 4, q1 = i % 4;
    int s1j = (q1 == 0) ? 0 : (4 * f1 + q1);
    int e1j = 1 + 4 * f1 + q1;
    for (int j = i + 1; j < 20; ++j) {
      int f2 = j / 4, q2 = j % 4;
      int s2j = (q2 == 0) ? 0 : (4 * f2 + q2);
      int e2j = 1 + 4 * f2 + q2;
      if (s1j == s2j || s1j == e2j || e1j == s2j) continue;
      t.s1[n] = (short)(41 * f1 + 10 * q1);
      t.s2[n] = (short)(41 * f2 + 10 * q2);
      ++n;
    }
  }
  return t;
}
__constant__ ComboTab dCombos = makeCombos();

// ---------- helpers ----------
__device__ __forceinline__ unsigned fflip(float f) {
  unsigned u = __float_as_uint(f);
  return (u >> 31) ? ~u : (u | 0x80000000u);
}
__device__ __forceinline__ float funflip(unsigned u) {
  return __uint_as_float((u >> 31) ? (u & 0x7FFFFFFFu) : ~u);
}
__device__ __forceinline__ void bone_joints(int p, int& sj, int& ej, float& tt) {
  int f = p / 41, r = p % 41;
  if (r == 40) { sj = ej = 4 + 4 * f; tt = 0.f; }
  else {
    int s = r / 10, k = r % 10;
    sj = (s == 0) ? 0 : (4 * f + s);
    ej = 1 + 4 * f + s;
    tt = (float)((double)k * 0.1);
  }
}
// min_dist[b][p] computed inline from flipped min + |p|^2
__device__ __forceinline__ float mind(const float* ws, int b, int p) {
  float s  = funflip(((const unsigned*)ws)[WS_MINS + b*208 + p]);
  float pp = ((const float4*)(ws + WS_POS4))[p].w;
  return sqrtf(fmaxf(s + pp, 1e-12f));
}

// ---------- K0a: small setup ----------
__global__ __launch_bounds__(256) void k0a_setup(const float* __restrict__ tpose,
                                                 const float* __restrict__ joint,
                                                 float* __restrict__ ws) {
  int t = threadIdx.x;
  float4* pos4 = (float4*)(ws + WS_POS4);
  for (int p = t; p < P_; p += 256) {
    int sj, ej; float tt; bone_joints(p, sj, ej, tt);
    float sx = tpose[sj*16+3], sy = tpose[sj*16+7], sz = tpose[sj*16+11];
    float ex = tpose[ej*16+3], ey = tpose[ej*16+7], ez = tpose[ej*16+11];
    float x = sx + (ex-sx)*tt, y = sy + (ey-sy)*tt, z = sz + (ez-sz)*tt;
    pos4[p] = make_float4(x, y, z, x*x + y*y + z*z);
  }
  float4* bone4 = (float4*)(ws + WS_BONE4);
  for (int i = t; i < B_*P_; i += 256) {
    int b = i / P_, p = i % P_;
    int sj, ej; float tt; bone_joints(p, sj, ej, tt);
    const float* js = joint + ((size_t)b*J_ + sj)*3;
    const float* je = joint + ((size_t)b*J_ + ej)*3;
    float x = js[0] + (je[0]-js[0])*tt;
    float y = js[1] + (je[1]-js[1])*tt;
    float z = js[2] + (je[2]-js[2])*tt;
    bone4[i] = make_float4(x, y, z, x*x + y*y + z*z);
  }
  unsigned* u = (unsigned*)ws;
  for (int i = t; i < 8*208; i += 256) u[WS_MINS + i] = 0xFFFFFFFFu;
  for (int i = t; i < 8*48;  i += 256) u[WS_NRS  + i] = 0xFFFFFFFFu;
  if (t < 8) ws[WS_RACC + t] = 0.f;
}

// ---------- K0b: pack WMMA A-operand buffers ----------
__global__ __launch_bounds__(256) void k0b_pack(const float* __restrict__ mesh,
                                                const float* __restrict__ geo,
                                                const int* __restrict__ seg,
                                                float* __restrict__ ws) {
  int i = blockIdx.x*256 + threadIdx.x;          // over B_*V_
  if (i >= B_*V_) return;
  int v = i & (V_-1);
  const float* m = mesh + (size_t)i*3;
  float mx = m[0], my = m[1], mz = m[2];
  ((float4*)(ws + WS_MESH4))[i] = make_float4(mx, my, mz, mx*mx + my*my + mz*mz);
  const float* g = geo + (size_t)i*3;
  float gx = g[0], gy = g[1], gz = g[2];
  float gg = gx*gx + gy*gy + gz*gz;
  float a3 = (seg[v] == 0) ? gg : 1e30f;         // poison masked rows; sqrt monotonic
  ((float4*)(ws + WS_GEO4))[i] = make_float4(gx, gy, gz, a3);
}

// ---------- WMMA min-scan core (packed A): load b64 -> wmma -> min3 ----------
__device__ __forceinline__ float wmma_min_scan(const float* __restrict__ abase,
                                               v2f bm, bool hi) {
  v8f cz = {0.f,0.f,0.f,0.f,0.f,0.f,0.f,0.f};
  v8f rm = {3e38f,3e38f,3e38f,3e38f,3e38f,3e38f,3e38f,3e38f};
  #pragma unroll 4
  for (int it = 0; it < 128; ++it) {
    v2f a = *(const v2f*)(abase + (size_t)it*64);   // 16 vertices * 4 floats
    v8f c = __builtin_amdgcn_wmma_f32_16x16x4_f32(false, a, false, bm,
                                                  (short)0, cz, false, false);
    #pragma unroll
    for (int q = 0; q < 8; ++q) rm[q] = fminf(rm[q], c[q]);
  }
  float m = rm[0];
  #pragma unroll
  for (int q = 1; q < 8; ++q) m = fminf(m, rm[q]);
  float o = __shfl_xor(m, 16, 32);   // rows 0-7 vs 8-15
  return fminf(m, o);
}

// ---------- K1 (packed): min over vertices of (|v|^2 - 2 v.p) ----------
__global__ __launch_bounds__(32) void k1_packed(float* __restrict__ ws) {
  int lane = threadIdx.x;
  int pt = blockIdx.x, b = blockIdx.y, ch = blockIdx.z;
  bool hi = lane >= 16;
  int n = lane & 15;
  int pc = pt*16 + n;
  int pcl = pc < P_ ? pc : (P_-1);
  float4 pp = ((const float4*)(ws + WS_POS4))[pcl];
  v2f bm;
  bm.x = hi ? (-2.f*pp.z) : (-2.f*pp.x);
  bm.y = hi ? 1.f         : (-2.f*pp.y);
  const float* abase = ws + WS_MESH4 + ((size_t)b*V_ + ch*2048 + n)*4 + (hi ? 2 : 0);
  float m = wmma_min_scan(abase, bm, hi);
  if (!hi && pc < P_)
    atomicMin(((unsigned*)ws) + WS_MINS + (b*208 + pc), fflip(m));
}

// ---------- K4 (packed): masked min for 40 non-rigid bone points ----------
__global__ __launch_bounds__(32) void k4_packed(float* __restrict__ ws) {
  int lane = threadIdx.x;
  int ct = blockIdx.x, b = blockIdx.y, ch = blockIdx.z;
  bool hi = lane >= 16;
  int n = lane & 15;
  int col = ct*16 + n;
  int cl = col < 40 ? col : 39;
  int e = cl / 10, k = cl % 10;
  int p = 41*(e+1) + 30 + k;
  float4 bn = ((const float4*)(ws + WS_BONE4))[b*P_ + p];
  v2f bm;
  bm.x = hi ? (-2.f*bn.z) : (-2.f*bn.x);
  bm.y = hi ? 1.f         : (-2.f*bn.y);
  const float* abase = ws + WS_GEO4 + ((size_t)b*V_ + ch*2048 + n)*4 + (hi ? 2 : 0);
  float m = wmma_min_scan(abase, bm, hi);
  if (!hi && col < 40)
    atomicMin(((unsigned*)ws) + WS_NRS + (b*48 + col), fflip(m));
}

// ---------- fallback (small ws): on-the-fly A construction ----------
__device__ __forceinline__ float wmma_min_scan_raw(const float* __restrict__ vbase,
                                                   const int* __restrict__ seg,
                                                   int n, v2f bm, bool hi, bool useSeg) {
  v8f cz = {0.f,0.f,0.f,0.f,0.f,0.f,0.f,0.f};
  v8f rm = {3e38f,3e38f,3e38f,3e38f,3e38f,3e38f,3e38f,3e38f};
  for (int it = 0; it < 128; ++it) {
    int v = it*16 + n;
    const float* r = vbase + (size_t)v*3;
    float x = r[0], y = r[1], z = r[2];
    float vv = x*x + y*y + z*z;
    if (useSeg) vv = (seg[v] == 0) ? vv : 1e30f;
    v2f a;
    a.x = hi ? z  : x;
    a.y = hi ? vv : y;
    v8f c = __builtin_amdgcn_wmma_f32_16x16x4_f32(false, a, false, bm,
                                                  (short)0, cz, false, false);
    #pragma unroll
    for (int q = 0; q < 8; ++q) rm[q] = fminf(rm[q], c[q]);
  }
  float m = rm[0];
  #pragma unroll
  for (int q = 1; q < 8; ++q) m = fminf(m, rm[q]);
  float o = __shfl_xor(m, 16, 32);
  return fminf(m, o);
}

__global__ __launch_bounds__(32) void k1_fb(const float* __restrict__ mesh,
                                            float* __restrict__ ws) {
  int lane = threadIdx.x;
  int pt = blockIdx.x, b = blockIdx.y, ch = blockIdx.z;
  bool hi = lane >= 16;
  int n = lane & 15;
  int pc = pt*16 + n;
  int pcl = pc < P_ ? pc : (P_-1);
  float4 pp = ((const float4*)(ws + WS_POS4))[pcl];
  v2f bm;
  bm.x = hi ? (-2.f*pp.z) : (-2.f*pp.x);
  bm.y = hi ? 1.f         : (-2.f*pp.y);
  const float* vbase = mesh + ((size_t)b*V_ + ch*2048)*3;
  float m = wmma_min_scan_raw(vbase, nullptr, n, bm, hi, false);
  if (!hi && pc < P_)
    atomicMin(((unsigned*)ws) + WS_MINS + (b*208 + pc), fflip(m));
}

__global__ __launch_bounds__(32) void k4_fb(const float* __restrict__ geo,
                                            const int* __restrict__ seg,
                                            float* __restrict__ ws) {
  int lane = threadIdx.x;
  int ct = blockIdx.x, b = blockIdx.y, ch = blockIdx.z;
  bool hi = lane >= 16;
  int n = lane & 15;
  int col = ct*16 + n;
  int cl = col < 40 ? col : 39;
  int e = cl / 10, k = cl % 10;
  int p = 41*(e+1) + 30 + k;
  float4 bn = ((const float4*)(ws + WS_BONE4))[b*P_ + p];
  v2f bm;
  bm.x = hi ? (-2.f*bn.z) : (-2.f*bn.x);
  bm.y = hi ? 1.f         : (-2.f*bn.y);
  const float* vbase = geo + ((size_t)b*V_ + ch*2048)*3;
  float m = wmma_min_scan_raw(vbase, seg + ch*2048, n, bm, hi, true);
  if (!hi && col < 40)
    atomicMin(((unsigned*)ws) + WS_NRS + (b*48 + col), fflip(m));
}

// ---------- K3: rigid term over 165 combos x 10x10 pairs ----------
__global__ __launch_bounds__(128) void k3_rigid(float* __restrict__ ws) {
  __shared__ float red[128];
  int t = threadIdx.x, c = blockIdx.x, b = blockIdx.y;
  float term = 0.f;
  if (t < 100) {
    int i = t / 10, j = t % 10;
    int p1 = dCombos.s1[c] + i;
    int p2 = dCombos.s2[c] + j;
    const float4* bo = (const float4*)(ws + WS_BONE4);
    float4 b1 = bo[b*P_ + p1], b2 = bo[b*P_ + p2];
    float dx = b1.x-b2.x, dy = b1.y-b2.y, dz = b1.z-b2.z;
    float d = sqrtf(dx*dx + dy*dy + dz*dz + 1e-12f);
    float thr = mind(ws, b, p1) + mind(ws, b, p2);
    term = fmaxf(thr - d, 0.f);
  }
  red[t] = term;
  __syncthreads();
  for (int s = 64; s > 0; s >>= 1) {
    if (t < s) red[t] += red[t + s];
    __syncthreads();
  }
  if (t == 0) atomicAdd(ws + WS_RACC + b, red[0] * 0.01f);
}

// ---------- K5: non-rigid finalize + combine output ----------
__global__ __launch_bounds__(64) void k5_final(float* __restrict__ ws,
                                               float* __restrict__ out) {
  __shared__ float cnr[32];
  int t = threadIdx.x;
  if (t < 32) {
    int b = t / 4, e = t % 4;
    float diffs[10];
    for (int k = 0; k < 10; ++k) {
      int col = e*10 + k;
      int p = 41*(e+1) + 30 + k;
      float s  = funflip(((unsigned*)ws)[WS_NRS + b*48 + col]);
      float bb = ((const float4*)(ws + WS_BONE4))[b*P_ + p].w;
      float dn = sqrtf(fmaxf(s + bb, 1e-12f));
      diffs[k] = dn - mind(ws, b, p);
    }
    bool has = false; int first = 0;
    for (int k = 0; k < 10; ++k) {
      if (diffs[k] < 0.f) { has = true; first = k; break; }
    }
    float contrib = 0.f;
    if (has) {
      float sum = 0.f; int cnt = 0;
      for (int k = first; k < 10; ++k) { sum += diffs[k]; ++cnt; }
      contrib = fabsf(sum / (float)cnt);
    }
    cnr[t] = contrib;
  }
  __syncthreads();
  if (t < 8) {
    float nr = cnr[t*4] + cnr[t*4+1] + cnr[t*4+2] + cnr[t*4+3];
    out[t] = ws[WS_RACC + t] / (float)NC_ + nr / 4.f;
  }
}

extern "C" void kernel_launch(void* const* d_in, const int* in_sizes, int n_in,
                              void* d_out, int out_size, void* d_ws, size_t ws_size,
                              hipStream_t stream) {
  const float* tpose = (const float*)d_in[0];   // (21,4,4)
  const float* mesh  = (const float*)d_in[1];   // (8,32768,3)
  const float* joint = (const float*)d_in[2];   // (8,21,3)
  const float* geo   = (const float*)d_in[3];   // (8,32768,3)
  const int*   seg   = (const int*)d_in[4];     // (32768,)
  float* ws  = (float*)d_ws;
  float* out = (float*)d_out;

  const bool packed = ws_size >= (size_t)WS_WORDS_PACKED * 4u;  // constant per run

  k0a_setup<<<1, 256, 0, stream>>>(tpose, joint, ws);
  if (packed) {
    k0b_pack<<<(B_*V_ + 255)/256, 256, 0, stream>>>(mesh, geo, seg, ws);
    k1_packed<<<dim3(13, 8, 16), 32, 0, stream>>>(ws);
    k4_packed<<<dim3(3, 8, 16), 32, 0, stream>>>(ws);
  } else {
    k1_fb<<<dim3(13, 8, 16), 32, 0, stream>>>(mesh, ws);
    k4_fb<<<dim3(3, 8, 16), 32, 0, stream>>>(geo, seg, ws);
  }
  k3_rigid<<<dim3(NC_, 8), 128, 0, stream>>>(ws);
  k5_final<<<1, 64, 0, stream>>>(ws, out);
}